// NeuralVolume_22857815949556
// MI455X (gfx1250) — compile-verified
//
#include <hip/hip_runtime.h>

#ifndef __has_builtin
#define __has_builtin(x) 0
#endif

// ---------------------------------------------------------------------------
// NeuralVolume dense multi-res trilinear encoder for gfx1250 (MI455X).
// Levels: R = 16, 32, 64, 128 ; C = 8 ; out = [B, 32] f32.
// Row offsets into embeddings: 0, 4096, 36864, 299008.
// Strategy: level-0 table (128 KB) -> LDS via async DMA (ASYNCcnt path);
// levels 1-3 gathered from L2-resident tables (76.7 MB total << 192 MB L2);
// outputs streamed with non-temporal stores to protect L2 table residency.
// ---------------------------------------------------------------------------

typedef float v4f __attribute__((ext_vector_type(4)));
typedef int   v4i __attribute__((ext_vector_type(4)));

typedef __attribute__((address_space(1))) v4i GV4;   // global int4
typedef __attribute__((address_space(3))) v4i LV4;   // LDS int4

__device__ __forceinline__ void async_g2l_b128(const void* g, void* l) {
#if __has_builtin(__builtin_amdgcn_global_load_async_to_lds_b128)
  __builtin_amdgcn_global_load_async_to_lds_b128(
      (GV4*)g, (LV4*)l, /*imm offset=*/0, /*cpol=*/0);
#else
  unsigned lofs = (unsigned)(__SIZE_TYPE__)(LV4*)l;   // low 32 bits = LDS byte addr
  asm volatile("global_load_async_to_lds_b128 %0, %1, off"
               :: "v"(lofs), "v"(g) : "memory");
#endif
}

__device__ __forceinline__ void wait_async_zero() {
#if __has_builtin(__builtin_amdgcn_s_wait_asynccnt)
  __builtin_amdgcn_s_wait_asynccnt(0);
#else
  asm volatile("s_wait_asynccnt 0" ::: "memory");
#endif
}

// Trilinear gather-accumulate for one level. R is a power of two (L2R = log2 R).
// Works for both LDS-backed (level 0) and global-backed tables via inlining +
// address-space inference.
template <int R, int L2R>
__device__ __forceinline__ void encode_level(float x, float y, float z,
                                             const float4* __restrict__ tab,
                                             float acc[8]) {
  const float s = (float)(R - 1);
  const float px = x * s, py = y * s, pz = z * s;
  const float bx = floorf(px), by = floorf(py), bz = floorf(pz);
  const float tx = px - bx, ty = py - by, tz = pz - bz;
  int ix = (int)bx, iy = (int)by, iz = (int)bz;
  ix = ix < 0 ? 0 : (ix > R - 2 ? R - 2 : ix);
  iy = iy < 0 ? 0 : (iy > R - 2 ? R - 2 : iy);
  iz = iz < 0 ? 0 : (iz > R - 2 ? R - 2 : iz);
  const float wx[2] = {1.0f - tx, tx};
  const float wy[2] = {1.0f - ty, ty};
  const float wz[2] = {1.0f - tz, tz};
#pragma unroll
  for (int c = 0; c < 8; ++c) {
    const int cx = c & 1, cy = (c >> 1) & 1, cz = (c >> 2) & 1;
    const unsigned idx = (unsigned)(ix + cx)
                       | ((unsigned)(iy + cy) << L2R)
                       | ((unsigned)(iz + cz) << (2 * L2R));
    const float w = wx[cx] * wy[cy] * wz[cz];
    const float4 r0 = tab[2u * idx + 0u];   // 32-B row = 2x b128
    const float4 r1 = tab[2u * idx + 1u];
    acc[0] = fmaf(w, r0.x, acc[0]);
    acc[1] = fmaf(w, r0.y, acc[1]);
    acc[2] = fmaf(w, r0.z, acc[2]);
    acc[3] = fmaf(w, r0.w, acc[3]);
    acc[4] = fmaf(w, r1.x, acc[4]);
    acc[5] = fmaf(w, r1.y, acc[5]);
    acc[6] = fmaf(w, r1.z, acc[6]);
    acc[7] = fmaf(w, r1.w, acc[7]);
  }
}

__device__ __forceinline__ void nt_store8(float* o, const float a[8]) {
  v4f v0 = {a[0], a[1], a[2], a[3]};
  v4f v1 = {a[4], a[5], a[6], a[7]};
  __builtin_nontemporal_store(v0, (v4f*)o);        // TH=NT: don't pollute L2
  __builtin_nontemporal_store(v1, (v4f*)(o + 4));
}

#define LVL0_ROWS 4096                 // 16^3
#define LVL0_F4   (LVL0_ROWS * 2)      // 8192 float4 = 128 KB

__global__ __launch_bounds__(256)
void neural_volume_encode(const float* __restrict__ inp,
                          const float* __restrict__ emb,
                          float* __restrict__ out,
                          int nPts) {
  // Level-0 table staged in LDS (128 KB of the 320 KB WGP pool).
  __shared__ float4 s_tab[LVL0_F4];

  // ---- async DMA fill: 8192 x b128, 32 per thread, tracked by ASYNCcnt ----
  {
    const char* g = (const char*)emb;        // level 0 starts at row 0
    char*       l = (char*)s_tab;
#pragma unroll
    for (int it = 0; it < 32; ++it) {
      const int ofs = ((int)threadIdx.x + it * 256) * 16;
      async_g2l_b128(g + ofs, l + ofs);
    }
  }
  wait_async_zero();    // my writes into LDS landed
  __syncthreads();      // everyone's writes landed

  const float4* __restrict__ tab1 = (const float4*)(emb + 4096ull   * 8);
  const float4* __restrict__ tab2 = (const float4*)(emb + 36864ull  * 8);
  const float4* __restrict__ tab3 = (const float4*)(emb + 299008ull * 8);

  const unsigned stride = gridDim.x * blockDim.x;
  for (unsigned i = blockIdx.x * blockDim.x + threadIdx.x; i < (unsigned)nPts;
       i += stride) {
    const float* p = inp + 3ull * i;
    const float x = __builtin_nontemporal_load(p + 0);
    const float y = __builtin_nontemporal_load(p + 1);
    const float z = __builtin_nontemporal_load(p + 2);

    float a0[8] = {0}, a1[8] = {0}, a2[8] = {0}, a3[8] = {0};
    encode_level<16, 4>(x, y, z, (const float4*)s_tab, a0);  // LDS gathers
    encode_level<32, 5>(x, y, z, tab1, a1);                  // L2-resident
    encode_level<64, 6>(x, y, z, tab2, a2);
    encode_level<128, 7>(x, y, z, tab3, a3);

    float* o = out + 32ull * i;
    nt_store8(o + 0,  a0);
    nt_store8(o + 8,  a1);
    nt_store8(o + 16, a2);
    nt_store8(o + 24, a3);
  }
}

extern "C" void kernel_launch(void* const* d_in, const int* in_sizes, int n_in,
                              void* d_out, int out_size, void* d_ws, size_t ws_size,
                              hipStream_t stream) {
  (void)n_in; (void)out_size; (void)d_ws; (void)ws_size;
  const float* inp = (const float*)d_in[0];   // [B,3] f32
  const float* emb = (const float*)d_in[1];   // [2396160, 8] f32
  float* out = (float*)d_out;                 // [B,32] f32
  const int nPts = in_sizes[0] / 3;

  // 512 blocks x 256 threads: grid-stride so each block amortizes its 128 KB
  // LDS fill over ~2048 points; 2 blocks per WGP (256/320 KB LDS).
  dim3 grid(512), block(256);
  hipLaunchKernelGGL(neural_volume_encode, grid, block, 0, stream,
                     inp, emb, out, nPts);
}